// QuantumLayer_695784702167
// MI455X (gfx1250) — compile-verified
//
#include <hip/hip_runtime.h>
#include <math.h>

typedef float v2f __attribute__((ext_vector_type(2)));
typedef float v8f __attribute__((ext_vector_type(8)));

#define NQ 4
#define NL 3
#define DIM 16

__device__ __forceinline__ float2 cmul(float2 a, float2 b) {
    return make_float2(a.x * b.x - a.y * b.y, a.x * b.y + a.y * b.x);
}
__device__ __forceinline__ float2 cadd(float2 a, float2 b) {
    return make_float2(a.x + b.x, a.y + b.y);
}

// ---------------------------------------------------------------------------
// Setup kernel (1 block, 32 threads, runs once): build the fixed 16x16 real
// symmetric matrix A = Re(W^H Z0 W), where W is the 3-layer entangling
// circuit unitary.  Thread j simulates basis state |j>.
// ---------------------------------------------------------------------------
__global__ void build_A_kernel(const float* __restrict__ w, float* __restrict__ A)
{
    __shared__ float2 W[DIM][DIM];   // W[k][j] = <k| W |j>
    const int j = threadIdx.x;

    if (j < DIM) {
        float2 st[DIM];
        #pragma unroll
        for (int k = 0; k < DIM; ++k) st[k] = make_float2(0.f, 0.f);
        st[j] = make_float2(1.f, 0.f);

        #pragma unroll
        for (int l = 0; l < NL; ++l) {
            // Rot(phi,theta,omega) = RZ(omega) RY(theta) RZ(phi) on each wire
            #pragma unroll
            for (int q = 0; q < NQ; ++q) {
                float phi   = w[(l * NQ + q) * 3 + 0];
                float theta = w[(l * NQ + q) * 3 + 1];
                float omega = w[(l * NQ + q) * 3 + 2];
                float ct = cosf(theta * 0.5f), sn = sinf(theta * 0.5f);
                float a = (phi + omega) * 0.5f, b = (phi - omega) * 0.5f;
                float2 ema = make_float2(cosf(a), -sinf(a));   // e^{-i a}
                float2 emb = make_float2(cosf(b), -sinf(b));   // e^{-i b}
                float2 U00 = make_float2( ema.x * ct,  ema.y * ct);
                float2 U01 = make_float2(-emb.x * sn,  emb.y * sn);   // -conj(emb)*s
                float2 U10 = make_float2( emb.x * sn,  emb.y * sn);
                float2 U11 = make_float2( ema.x * ct, -ema.y * ct);   // conj(ema)*c
                const int mask = 1 << (3 - q);                 // qubit 0 = MSB
                #pragma unroll
                for (int k = 0; k < DIM; ++k) {
                    if ((k & mask) == 0) {
                        int k1 = k | mask;
                        float2 v0 = st[k], v1 = st[k1];
                        st[k]  = cadd(cmul(U00, v0), cmul(U01, v1));
                        st[k1] = cadd(cmul(U10, v0), cmul(U11, v1));
                    }
                }
            }
            // ring of CNOTs, range r = l%3 + 1, order q = 0..3
            const int r = l % (NQ - 1) + 1;
            #pragma unroll
            for (int q = 0; q < NQ; ++q) {
                const int cm = 1 << (3 - q);
                const int tm = 1 << (3 - ((q + r) % NQ));
                #pragma unroll
                for (int k = 0; k < DIM; ++k) {
                    if ((k & cm) && !(k & tm)) {
                        int k1 = k | tm;
                        float2 tmp = st[k]; st[k] = st[k1]; st[k1] = tmp;
                    }
                }
            }
        }
        #pragma unroll
        for (int k = 0; k < DIM; ++k) W[k][j] = st[k];
    }
    __syncthreads();
    if (j < DIM) {
        // A[i][j] = sum_k z_k * Re( conj(W[k][i]) * W[k][j] ),  z_k = +1 if bit(q0)=0
        for (int i = 0; i < DIM; ++i) {
            float acc = 0.f;
            #pragma unroll
            for (int k = 0; k < DIM; ++k) {
                float zk = (k & 8) ? -1.f : 1.f;
                float2 wki = W[k][i], wkj = W[k][j];
                acc += zk * (wki.x * wkj.x + wki.y * wkj.y);
            }
            A[i * DIM + j] = acc;
        }
    }
}

// ---------------------------------------------------------------------------
// Main kernel: each wave handles tiles of 16 samples.
//   Psi: 16x16 (element k of sample n), real.
//   Y = A * Psi via 4x V_WMMA_F32_16X16X4_F32 (K = 16 in chunks of 4)
//   z_n = sum_m Psi[m][n] * Y[m][n]  (C/D-layout dot + xor-16 shuffle)
// ---------------------------------------------------------------------------
__global__ void __launch_bounds__(256)
qcircuit_wmma_kernel(const float* __restrict__ x, const float* __restrict__ A,
                     float* __restrict__ out, int nTiles, int tilesPerWave)
{
    const int lane   = threadIdx.x & 31;
    const int waveId = (int)((blockIdx.x * blockDim.x + threadIdx.x) >> 5);
    const bool hi    = lane >= 16;     // upper half-wave
    const int  col   = lane & 15;

    // A-matrix fragments (16x4 f32 per chunk): lane row = col,
    // lanes 0-15 hold K = 4t+0,4t+1 ; lanes 16-31 hold K = 4t+2,4t+3
    v2f aFrag[4];
    #pragma unroll
    for (int t = 0; t < 4; ++t) {
        int k0 = 4 * t + (hi ? 2 : 0);
        aFrag[t][0] = A[col * DIM + k0];
        aFrag[t][1] = A[col * DIM + k0 + 1];
    }

    for (int it = 0; it < tilesPerWave; ++it) {
        const int tile = waveId * tilesPerWave + it;   // wave-uniform
        if (tile >= nTiles) return;                    // uniform exit: EXEC stays all-1 for WMMA
        const int sample = tile * 16 + col;

        const float4 xv = *(const float4*)(x + 4 * (long)sample);
        float c0, s0, c1, s1, c2, s2, c3, s3;
        sincosf(xv.x * 0.5f, &s0, &c0);
        sincosf(xv.y * 0.5f, &s1, &c1);
        sincosf(xv.z * 0.5f, &s2, &c2);
        sincosf(xv.w * 0.5f, &s3, &c3);

        // psi_k = f0[k>>3] * f1[(k>>2)&1] * f2[(k>>1)&1] * f3[k&1], f(0)=cos, f(1)=sin
        // B-fragment needs k = 4t + 2*hi + {0,1}  ->  bit1 = hi, bits3:2 = t
        const float w2   = hi ? s2 : c2;
        const float w2c3 = w2 * c3;
        const float w2s3 = w2 * s3;
        const float g[4] = { c0 * c1, c0 * s1, s0 * c1, s0 * s1 };   // f0[t>>1]*f1[t&1]

        v8f acc = {0.f, 0.f, 0.f, 0.f, 0.f, 0.f, 0.f, 0.f};
        #pragma unroll
        for (int t = 0; t < 4; ++t) {
            v2f bFrag;
            bFrag[0] = g[t] * w2c3;
            bFrag[1] = g[t] * w2s3;
            acc = __builtin_amdgcn_wmma_f32_16x16x4_f32(
                /*neg_a=*/false, aFrag[t], /*neg_b=*/false, bFrag,
                /*c_mod=*/(short)0, acc, /*reuse_a=*/false, /*reuse_b=*/false);
        }

        // C/D layout: lane holds rows m = 8*hi + r, r = 0..7.
        // psi_m = w0 * f1[r>>2] * f2[(r>>1)&1] * f3[r&1]
        const float w0 = hi ? s0 : c0;
        const float p[8] = { c1 * c2 * c3, c1 * c2 * s3, c1 * s2 * c3, c1 * s2 * s3,
                             s1 * c2 * c3, s1 * c2 * s3, s1 * s2 * c3, s1 * s2 * s3 };
        float dot = 0.f;
        #pragma unroll
        for (int r = 0; r < 8; ++r)
            dot += (w0 * p[r]) * acc[r];

        const float z = dot + __shfl_xor(dot, 16, 32);  // combine m=0..7 and m=8..15 halves
        if (!hi) out[sample] = z;
    }
}

extern "C" void kernel_launch(void* const* d_in, const int* in_sizes, int n_in,
                              void* d_out, int out_size, void* d_ws, size_t ws_size,
                              hipStream_t stream) {
    const float* x = (const float*)d_in[0];     // (B, 4) float32
    const float* w = (const float*)d_in[1];     // (3, 4, 3) float32
    float* out = (float*)d_out;                 // (B, 1) float32
    float* A   = (float*)d_ws;                  // 16x16 float32 scratch

    const int B = in_sizes[0] / NQ;

    build_A_kernel<<<1, 32, 0, stream>>>(w, A);

    const int nTiles = B / 16;                  // B = 524288 -> 32768 tiles
    const int tilesPerWave = 4;
    const int waves = (nTiles + tilesPerWave - 1) / tilesPerWave;
    const int threads = 256;                    // 8 waves/block
    const int blocks = (waves + (threads / 32) - 1) / (threads / 32);

    qcircuit_wmma_kernel<<<blocks, threads, 0, stream>>>(x, A, out, nTiles, tilesPerWave);
}